// KeywordEncoder_9680856285391
// MI455X (gfx1250) — compile-verified
//
#include <hip/hip_runtime.h>

typedef __attribute__((ext_vector_type(2))) float v2f;
typedef __attribute__((ext_vector_type(8))) float v8f;

#define BAGS_PER_BLOCK 32
#define L_MAX 50
#define EMBED 64
#define COND 256
#define A_STRIDE 68   // 64 + 4 pad floats; row = 272B (16B multiple) -> aligned b128 LDS stores,
                      // and A-fragment ds_load_b64 reads touch all 64 banks exactly once.

__global__ __launch_bounds__(256)
void keyword_encoder_kernel(const int* __restrict__ token_ids,
                            const int* __restrict__ lengths,
                            const float* __restrict__ emb,
                            const float* __restrict__ W,
                            const float* __restrict__ bias,
                            const float* __restrict__ null_emb,
                            float* __restrict__ out)
{
    __shared__ __align__(16) float sA[BAGS_PER_BLOCK * A_STRIDE];
    __shared__ int s_ids[BAGS_PER_BLOCK * L_MAX];
    __shared__ int s_len[BAGS_PER_BLOCK];

    const int tid     = threadIdx.x;
    const int bagBase = blockIdx.x * BAGS_PER_BLOCK;

    // WMMA-phase coordinates (computed early so we can prefetch W rows now)
    const int wave = tid >> 5;
    const int lane = tid & 31;
    const int lo   = lane & 15;
    const int hi   = lane >> 4;
    const int n0   = wave * 2, n1 = n0 + 1;
    const int c0   = n0 * 16 + lo;
    const int c1   = n1 * 16 + lo;

    // Warm the weight rows this wave will consume (global_prefetch_b8).
    __builtin_prefetch(W + (size_t)c0 * EMBED, 0, 0);
    __builtin_prefetch(W + (size_t)c1 * EMBED, 0, 0);

    // ---- Stage token ids & lengths into LDS (coalesced) ----
    for (int i = tid; i < BAGS_PER_BLOCK * L_MAX; i += 256)
        s_ids[i] = token_ids[bagBase * L_MAX + i];
    if (tid < BAGS_PER_BLOCK)
        s_len[tid] = lengths[bagBase + tid];
    __syncthreads();

    // ---- Phase 1: masked-mean gather. 16 lanes per bag; each lane loads float4
    //      so a half-wave fetches one full 256B embedding row (global_load_b128). ----
    const int g   = tid >> 4;   // 0..15: row group
    const int l16 = tid & 15;
    #pragma unroll
    for (int rep = 0; rep < 2; ++rep) {
        const int bagl = g + rep * 16;          // local bag 0..31
        const int len  = s_len[bagl];
        const int* ids = &s_ids[bagl * L_MAX];
        float4 acc = make_float4(0.f, 0.f, 0.f, 0.f);
        for (int t = 0; t < len; ++t) {
            const float4 r = *(const float4*)(emb + (size_t)ids[t] * EMBED + l16 * 4);
            acc.x += r.x; acc.y += r.y; acc.z += r.z; acc.w += r.w;
        }
        const float scale = 1.0f / (float)((len > 0) ? len : 1);
        float4 m = make_float4(acc.x * scale, acc.y * scale, acc.z * scale, acc.w * scale);
        *(float4*)&sA[bagl * A_STRIDE + l16 * 4] = m;
    }
    __syncthreads();

    // ---- Phase 2: 32x256 output tile = (2 M-tiles) x (16 N-tiles).
    //      Wave w owns N-tiles {2w, 2w+1}; K=64 in steps of 4 via v_wmma_f32_16x16x4_f32.
    //      A-frag layout (ISA 16x4 fp32): lane -> M = lane%16, K = 2*(lane/16) + vgpr.
    //      B-frag (4x16):                 lane -> N = lane%16, K = 2*(lane/16) + vgpr.  ----
    v8f acc00 = {}, acc01 = {}, acc10 = {}, acc11 = {};

    const float* w0p = W + (size_t)c0 * EMBED + hi * 2;
    const float* w1p = W + (size_t)c1 * EMBED + hi * 2;
    const float* a0p = &sA[lo * A_STRIDE + hi * 2];
    const float* a1p = &sA[(lo + 16) * A_STRIDE + hi * 2];

    #pragma unroll
    for (int k = 0; k < EMBED / 4; ++k) {
        const v2f a0 = *(const v2f*)(a0p + k * 4);
        const v2f a1 = *(const v2f*)(a1p + k * 4);
        const v2f b0 = *(const v2f*)(w0p + k * 4);
        const v2f b1 = *(const v2f*)(w1p + k * 4);
        acc00 = __builtin_amdgcn_wmma_f32_16x16x4_f32(false, a0, false, b0, (short)0, acc00, false, false);
        acc01 = __builtin_amdgcn_wmma_f32_16x16x4_f32(false, a0, false, b1, (short)0, acc01, false, false);
        acc10 = __builtin_amdgcn_wmma_f32_16x16x4_f32(false, a1, false, b0, (short)0, acc10, false, false);
        acc11 = __builtin_amdgcn_wmma_f32_16x16x4_f32(false, a1, false, b1, (short)0, acc11, false, false);
    }

    // ---- Epilogue: bias add + empty-bag null override, then store.
    //      C layout: vgpr v, lanes 0-15 -> M=v, N=lane; lanes 16-31 -> M=v+8. ----
    const float bias0 = bias[c0], bias1 = bias[c1];
    const float nul0  = null_emb[c0], nul1 = null_emb[c1];

    #pragma unroll
    for (int v = 0; v < 8; ++v) {
        const int m0 = v + 8 * hi;       // row within M-tile 0
        const int m1 = 16 + m0;          // row within M-tile 1
        const bool e0 = (s_len[m0] == 0);
        const bool e1 = (s_len[m1] == 0);
        float* r0 = out + (size_t)(bagBase + m0) * COND;
        float* r1 = out + (size_t)(bagBase + m1) * COND;
        r0[c0] = e0 ? nul0 : (acc00[v] + bias0);
        r0[c1] = e0 ? nul1 : (acc01[v] + bias1);
        r1[c0] = e1 ? nul0 : (acc10[v] + bias0);
        r1[c1] = e1 ? nul1 : (acc11[v] + bias1);
    }
}

extern "C" void kernel_launch(void* const* d_in, const int* in_sizes, int n_in,
                              void* d_out, int out_size, void* d_ws, size_t ws_size,
                              hipStream_t stream) {
    const int*   token_ids = (const int*)  d_in[0];   // [B, L]
    const int*   lengths   = (const int*)  d_in[1];   // [B]
    const float* emb       = (const float*)d_in[2];   // [VOCAB, 64]
    const float* W         = (const float*)d_in[3];   // [256, 64]
    const float* b         = (const float*)d_in[4];   // [256]
    const float* null_emb  = (const float*)d_in[5];   // [256]
    float*       out       = (float*)d_out;           // [B, 256]

    const int B = in_sizes[1];                        // lengths element count
    dim3 grid(B / BAGS_PER_BLOCK), block(256);
    hipLaunchKernelGGL(keyword_encoder_kernel, grid, block, 0, stream,
                       token_ids, lengths, emb, W, b, null_emb, out);
}